// MacroTopologyGNN_89429809037954
// MI455X (gfx1250) — compile-verified
//
#include <hip/hip_runtime.h>
#include <math.h>

#define D 128
#define Hh 4
#define Gg 64
#define NEG_SLOPE 0.2f
#define EPSN 1e-5f

typedef __attribute__((ext_vector_type(2))) float v2f;
typedef __attribute__((ext_vector_type(8))) float v8f;

// ---- monotone float <-> uint encoding for atomicMax on signed floats ----
__device__ __forceinline__ unsigned enc_f(float f) {
  unsigned u = __float_as_uint(f);
  return (u & 0x80000000u) ? ~u : (u | 0x80000000u);
}
__device__ __forceinline__ float dec_f(unsigned u) {
  unsigned v = (u & 0x80000000u) ? (u ^ 0x80000000u) : ~u;
  return __uint_as_float(v);
}

// =====================  WMMA GEMM: Y = X @ W + bias  =====================
// X: n x 128, W: 128 x 128 (row major), Y: n x 128.
// Block = 256 threads = 8 waves. Each wave: one 16-row M tile, all 8 N tiles.
// W staged in LDS (64KB of the 320KB WGP LDS).
__global__ __launch_bounds__(256) void gemm_wmma(const float* __restrict__ X,
                                                 const float* __restrict__ W,
                                                 const float* __restrict__ bias,
                                                 float* __restrict__ Y, int n) {
  __shared__ float wsh[D * D];  // 64 KB
  {
    const float4* Wv = (const float4*)W;
    float4* Sv = (float4*)wsh;
    for (int j = threadIdx.x; j < (D * D) / 4; j += blockDim.x) Sv[j] = Wv[j];
  }
  __syncthreads();

  const int wave  = threadIdx.x >> 5;
  const int lane  = threadIdx.x & 31;
  const int mlo   = lane & 15;   // M index (A) / N index (B,C,D)
  const int khalf = lane >> 4;   // selects K pair {0,1} vs {2,3}; M row +8 for C/D
  const int m0 = blockIdx.x * 128 + wave * 16;
  if (m0 >= n) return;  // wave-uniform: EXEC stays all-1s for active waves

  const float* xrow = X + (size_t)(m0 + mlo) * D;

  for (int nt = 0; nt < 8; ++nt) {
    const int n0 = nt * 16;
    v8f acc = {};
#pragma unroll
    for (int k = 0; k < D; k += 4) {
      // A fragment: 16x4 f32, lane=M, VGPR0/1 = K = khalf*2 + {0,1}
      float2 av = *(const float2*)(xrow + k + khalf * 2);
      v2f a; a[0] = av.x; a[1] = av.y;
      // B fragment: 4x16 f32, lane=N, VGPR0/1 = K = khalf*2 + {0,1}
      const int bi = (k + khalf * 2) * D + n0 + mlo;
      v2f b; b[0] = wsh[bi]; b[1] = wsh[bi + D];
      acc = __builtin_amdgcn_wmma_f32_16x16x4_f32(false, a, false, b,
                                                  (short)0, acc, false, false);
    }
    const float bb = bias[n0 + mlo];
#pragma unroll
    for (int r = 0; r < 8; ++r) {
      const int row = m0 + r + khalf * 8;  // C/D layout: VGPR r -> M=r (+8 hi half)
      Y[(size_t)row * D + n0 + mlo] = acc[r] + bb;
    }
  }
}

// =====================  init: hbuf = x + bias, zero accumulators  ========
__global__ void init_kernel(const float* __restrict__ x,
                            const float* __restrict__ bias,
                            float* __restrict__ hbuf, unsigned* __restrict__ maxkey,
                            float* __restrict__ denom, float* __restrict__ gsum,
                            float* __restrict__ vsum, float* __restrict__ cnt, int n) {
  long i = (long)blockIdx.x * blockDim.x + threadIdx.x;
  long total = (long)n * D;
  if (i < total) hbuf[i] = x[i] + bias[i & (D - 1)];
  if (i < (long)n * Hh) { maxkey[i] = 0u; denom[i] = 0.f; }
  if (i < Gg * D) { gsum[i] = 0.f; vsum[i] = 0.f; }
  if (i < Gg) cnt[i] = 0.f;
}

// =====================  score + segment max (wave per edge)  =============
__global__ __launch_bounds__(256) void score_kernel(
    const float* __restrict__ xl, const float* __restrict__ xr,
    const float* __restrict__ att, const int* __restrict__ ei,
    float* __restrict__ score, unsigned* __restrict__ maxkey, int nE, int nN) {
  int wid = (int)(((long)blockIdx.x * blockDim.x + threadIdx.x) >> 5);
  int lane = threadIdx.x & 31;
  int etot = nE + nN;
  if (wid >= etot) return;
  int s, d;
  if (wid < nE) { s = ei[wid]; d = ei[nE + wid]; } else { s = wid - nE; d = s; }
  int c0 = lane * 4;  // lanes 0-7 head 0, 8-15 head 1, ...
  float4 xlv = *(const float4*)(xl + (size_t)s * D + c0);
  float4 xrv = *(const float4*)(xr + (size_t)d * D + c0);
  float4 av  = *(const float4*)(att + c0);
  float p = 0.f, e;
  e = xlv.x + xrv.x; e = e > 0.f ? e : NEG_SLOPE * e; p += e * av.x;
  e = xlv.y + xrv.y; e = e > 0.f ? e : NEG_SLOPE * e; p += e * av.y;
  e = xlv.z + xrv.z; e = e > 0.f ? e : NEG_SLOPE * e; p += e * av.z;
  e = xlv.w + xrv.w; e = e > 0.f ? e : NEG_SLOPE * e; p += e * av.w;
  // reduce within each 8-lane head group (wave32)
  p += __shfl_xor(p, 1, 32);
  p += __shfl_xor(p, 2, 32);
  p += __shfl_xor(p, 4, 32);
  if ((lane & 7) == 0) {
    int h = lane >> 3;
    score[(size_t)wid * Hh + h] = p;
    atomicMax(&maxkey[(size_t)d * Hh + h], enc_f(p));
  }
}

// =====================  exp(score - m[dst]) and denom accumulation  ======
__global__ void expden_kernel(const int* __restrict__ ei, float* __restrict__ score,
                              const unsigned* __restrict__ maxkey,
                              float* __restrict__ denom, int nE, int nN) {
  long i = (long)blockIdx.x * blockDim.x + threadIdx.x;
  long total = (long)(nE + nN) * Hh;
  if (i >= total) return;
  int eidx = (int)(i >> 2);
  int h = (int)(i & 3);
  int d = (eidx < nE) ? ei[nE + eidx] : (eidx - nE);
  float m = dec_f(maxkey[(size_t)d * Hh + h]);
  float a = expf(score[i] - m);
  score[i] = a;  // overwrite with numerator
  atomicAdd(&denom[(size_t)d * Hh + h], a);
}

// =====================  aggregate: hbuf[dst] += alpha * xl[src]  =========
__global__ __launch_bounds__(256) void aggregate_kernel(
    const float* __restrict__ xl, const float* __restrict__ score,
    const float* __restrict__ denom, const int* __restrict__ ei,
    float* __restrict__ hbuf, int nE, int nN) {
  int wid = (int)(((long)blockIdx.x * blockDim.x + threadIdx.x) >> 5);
  int lane = threadIdx.x & 31;
  int etot = nE + nN;
  if (wid >= etot) return;
  int s, d;
  if (wid < nE) { s = ei[wid]; d = ei[nE + wid]; } else { s = wid - nE; d = s; }
  int h = lane >> 3;
  float a   = score[(size_t)wid * Hh + h];
  float den = denom[(size_t)d * Hh + h];
  float alpha = a / fmaxf(den, 1e-16f);
  int c0 = lane * 4;
  float4 xlv = *(const float4*)(xl + (size_t)s * D + c0);
  float* hp = hbuf + (size_t)d * D + c0;
  atomicAdd(hp + 0, alpha * xlv.x);
  atomicAdd(hp + 1, alpha * xlv.y);
  atomicAdd(hp + 2, alpha * xlv.z);
  atomicAdd(hp + 3, alpha * xlv.w);
}

// =====================  GraphNorm stats: sum + count  ====================
__global__ __launch_bounds__(256) void gsum_kernel(const float* __restrict__ hbuf,
                                                   const int* __restrict__ batch,
                                                   float* __restrict__ gsum,
                                                   float* __restrict__ cnt, int n) {
  int nid = (int)(((long)blockIdx.x * blockDim.x + threadIdx.x) >> 5);
  int lane = threadIdx.x & 31;
  if (nid >= n) return;
  int g = batch[nid];
  int c0 = lane * 4;
  float4 hv = *(const float4*)(hbuf + (size_t)nid * D + c0);
  float* gp = gsum + (size_t)g * D + c0;
  atomicAdd(gp + 0, hv.x); atomicAdd(gp + 1, hv.y);
  atomicAdd(gp + 2, hv.z); atomicAdd(gp + 3, hv.w);
  if (lane == 0) atomicAdd(&cnt[g], 1.0f);
}

__global__ void mean_kernel(float* __restrict__ gsum, const float* __restrict__ cnt) {
  int i = blockIdx.x * blockDim.x + threadIdx.x;
  if (i >= Gg * D) return;
  gsum[i] = gsum[i] / fmaxf(cnt[i >> 7], 1.0f);  // becomes mean
}

// hc = h - gn_mean_scale * mean[g]; hbuf <- hc; vsum[g] += hc*hc
__global__ __launch_bounds__(256) void center_kernel(
    float* __restrict__ hbuf, const int* __restrict__ batch,
    const float* __restrict__ mean, const float* __restrict__ mscale,
    float* __restrict__ vsum, int n) {
  int nid = (int)(((long)blockIdx.x * blockDim.x + threadIdx.x) >> 5);
  int lane = threadIdx.x & 31;
  if (nid >= n) return;
  int g = batch[nid];
  int c0 = lane * 4;
  float4 hv = *(const float4*)(hbuf + (size_t)nid * D + c0);
  float4 mv = *(const float4*)(mean + (size_t)g * D + c0);
  float4 sv = *(const float4*)(mscale + c0);
  float4 hc;
  hc.x = hv.x - sv.x * mv.x;
  hc.y = hv.y - sv.y * mv.y;
  hc.z = hv.z - sv.z * mv.z;
  hc.w = hv.w - sv.w * mv.w;
  *(float4*)(hbuf + (size_t)nid * D + c0) = hc;
  float* vp = vsum + (size_t)g * D + c0;
  atomicAdd(vp + 0, hc.x * hc.x); atomicAdd(vp + 1, hc.y * hc.y);
  atomicAdd(vp + 2, hc.z * hc.z); atomicAdd(vp + 3, hc.w * hc.w);
}

__global__ void inv_kernel(float* __restrict__ vsum, const float* __restrict__ cnt) {
  int i = blockIdx.x * blockDim.x + threadIdx.x;
  if (i >= Gg * D) return;
  float c = fmaxf(cnt[i >> 7], 1.0f);
  vsum[i] = 1.0f / sqrtf(vsum[i] / c + EPSN);  // becomes 1/sqrt(var+eps)
}

// out = gelu(gn_weight * hc * inv[g] + gn_bias)   (exact gelu, erf form)
__global__ void final_kernel(const float* __restrict__ hbuf,
                             const int* __restrict__ batch,
                             const float* __restrict__ inv,
                             const float* __restrict__ w,
                             const float* __restrict__ b,
                             float* __restrict__ out, int n) {
  long i = (long)blockIdx.x * blockDim.x + threadIdx.x;
  if (i >= (long)n * D) return;
  int c = (int)(i & (D - 1));
  int g = batch[i >> 7];
  float hn = w[c] * hbuf[i] * inv[(size_t)g * D + c] + b[c];
  out[i] = 0.5f * hn * (1.0f + erff(hn * 0.70710678118654752f));
}

extern "C" void kernel_launch(void* const* d_in, const int* in_sizes, int n_in,
                              void* d_out, int out_size, void* d_ws, size_t ws_size,
                              hipStream_t stream) {
  const float* x    = (const float*)d_in[0];
  const float* Wl   = (const float*)d_in[1];
  const float* bl   = (const float*)d_in[2];
  const float* Wr   = (const float*)d_in[3];
  const float* br   = (const float*)d_in[4];
  const float* att  = (const float*)d_in[5];
  const float* bias = (const float*)d_in[6];
  const float* gw   = (const float*)d_in[7];
  const float* gb   = (const float*)d_in[8];
  const float* gms  = (const float*)d_in[9];
  const int*   ei   = (const int*)d_in[10];
  const int*   batch= (const int*)d_in[11];
  float* out = (float*)d_out;

  const int n    = in_sizes[0] / D;
  const int nE   = in_sizes[10] / 2;
  const int etot = nE + n;

  char* ws = (char*)d_ws;
  size_t off = 0;
  float*    xl     = (float*)(ws + off); off += (size_t)n * D * 4;
  float*    xr     = (float*)(ws + off); off += (size_t)n * D * 4;
  float*    hbuf   = (float*)(ws + off); off += (size_t)n * D * 4;
  float*    score  = (float*)(ws + off); off += (size_t)etot * Hh * 4;
  unsigned* maxkey = (unsigned*)(ws + off); off += (size_t)n * Hh * 4;
  float*    denom  = (float*)(ws + off); off += (size_t)n * Hh * 4;
  float*    gsum   = (float*)(ws + off); off += (size_t)Gg * D * 4;
  float*    vsum   = (float*)(ws + off); off += (size_t)Gg * D * 4;
  float*    cnt    = (float*)(ws + off); off += (size_t)Gg * 4;

  const int gemmBlocks  = (n + 127) / 128;
  gemm_wmma<<<gemmBlocks, 256, 0, stream>>>(x, Wl, bl, xl, n);
  gemm_wmma<<<gemmBlocks, 256, 0, stream>>>(x, Wr, br, xr, n);

  const long totND = (long)n * D;
  init_kernel<<<(int)((totND + 255) / 256), 256, 0, stream>>>(
      x, bias, hbuf, maxkey, denom, gsum, vsum, cnt, n);

  const int edgeBlocks = (etot + 7) / 8;  // 8 waves per 256-thread block
  score_kernel<<<edgeBlocks, 256, 0, stream>>>(xl, xr, att, ei, score, maxkey, nE, n);

  const long totEH = (long)etot * Hh;
  expden_kernel<<<(int)((totEH + 255) / 256), 256, 0, stream>>>(
      ei, score, maxkey, denom, nE, n);

  aggregate_kernel<<<edgeBlocks, 256, 0, stream>>>(xl, score, denom, ei, hbuf, nE, n);

  const int nodeBlocks = (n + 7) / 8;
  gsum_kernel<<<nodeBlocks, 256, 0, stream>>>(hbuf, batch, gsum, cnt, n);
  mean_kernel<<<(Gg * D + 255) / 256, 256, 0, stream>>>(gsum, cnt);
  center_kernel<<<nodeBlocks, 256, 0, stream>>>(hbuf, batch, gsum, gms, vsum, n);
  inv_kernel<<<(Gg * D + 255) / 256, 256, 0, stream>>>(vsum, cnt);
  final_kernel<<<(int)((totND + 255) / 256), 256, 0, stream>>>(
      hbuf, batch, vsum, gw, gb, out, n);
}